// Router_34720515621320
// MI455X (gfx1250) — compile-verified
//
#include <hip/hip_runtime.h>

typedef __attribute__((ext_vector_type(2))) float v2f;
typedef __attribute__((ext_vector_type(4))) float v4f;
typedef __attribute__((ext_vector_type(8))) float v8f;

// ---------------------------------------------------------------------------
// Kernel 1: streaming zero-fill of the 536 MB output (the bandwidth bottleneck).
// Nontemporal b128 stores so we don't thrash L2 with write-allocate traffic.
// ---------------------------------------------------------------------------
__global__ __launch_bounds__(256)
void router_zero_fill(float* __restrict__ out, size_t n) {
  const size_t tid = (size_t)blockIdx.x * blockDim.x + threadIdx.x;
  const size_t stride = (size_t)gridDim.x * blockDim.x;
  const v4f z = {0.f, 0.f, 0.f, 0.f};
  const size_t n4 = n >> 2;
  for (size_t i = tid; i < n4; i += stride)
    __builtin_nontemporal_store(z, ((v4f*)out) + i);
  const size_t t = (n4 << 2) + tid;
  if (t < n) out[t] = 0.0f;
}

// ---------------------------------------------------------------------------
// Kernel 2: gating GEMM (fp32 WMMA 16x16x4) + per-token top-2 + softmax(2).
// One wave per 16-token tile. A = x[16 tokens, 4 k], lane layout per ISA:
//   lanes 0-31: M = lane&15; VGPR0/1 hold K = 2*(lane>>4) + {0,1}.
// B = w_g^T (4x16): lane = N column; only N<8 matters for D, so lanes 8-15
// duplicate rows 0-7 (their D columns are never read -> no masking needed).
// ---------------------------------------------------------------------------
__global__ __launch_bounds__(32)
void router_gate_top2(const float* __restrict__ x, const float* __restrict__ wg,
                      int* __restrict__ tk_idx, float* __restrict__ tk_p, int H) {
  __shared__ float logits_lds[16 * 8];
  const int lane = threadIdx.x;
  const int half = lane >> 4;   // 0 or 1
  const int l15  = lane & 15;
  const int tile = blockIdx.x;  // 16 tokens per tile

  const float* arow = x  + (size_t)(tile * 16 + l15) * H + 2 * half;
  const float* brow = wg + (size_t)(l15 & 7) * H + 2 * half;

  v8f acc = {0.f, 0.f, 0.f, 0.f, 0.f, 0.f, 0.f, 0.f};
  for (int k0 = 0; k0 < H; k0 += 4) {
    v2f a = *(const v2f*)(arow + k0);
    v2f b = *(const v2f*)(brow + k0);
    // (neg_a, A, neg_b, B, c_mod, C, reuse_a, reuse_b)
    acc = __builtin_amdgcn_wmma_f32_16x16x4_f32(false, a, false, b,
                                                (short)0, acc, false, false);
  }

  // D layout: VGPR r in lane l holds logits[M = r + 8*half][N = l&15].
  if (l15 < 8) {
#pragma unroll
    for (int r = 0; r < 8; ++r)
      logits_lds[(r + 8 * half) * 8 + l15] = acc[r];
  }
  __syncthreads();

  if (lane < 16) {
    const int m = lane;
    float best_v = -__builtin_inff(); int best_e = 0;
#pragma unroll
    for (int e = 0; e < 8; ++e) {
      float v = logits_lds[m * 8 + e];
      if (v > best_v) { best_v = v; best_e = e; }   // strict > : lower idx wins ties
    }
    float sec_v = -__builtin_inff(); int sec_e = 0;
#pragma unroll
    for (int e = 0; e < 8; ++e) {
      float v = logits_lds[m * 8 + e];
      if (e != best_e && v > sec_v) { sec_v = v; sec_e = e; }
    }
    const int n = tile * 16 + m;
    // softmax over the two kept logits (-inf elsewhere): p1 = 1/(1+e2), p2 = e2/(1+e2)
    const float e2  = expf(sec_v - best_v);
    const float inv = 1.0f / (1.0f + e2);
    tk_idx[2 * n + 0] = best_e;
    tk_idx[2 * n + 1] = sec_e;
    tk_p  [2 * n + 0] = inv;
    tk_p  [2 * n + 1] = e2 * inv;
  }
}

// ---------------------------------------------------------------------------
// Kernel 3: sequential capacity ranking (cumsum in k-major, token order) +
// scatter. One wave32; per-expert intra-wave prefix via ballot/popcount,
// running per-expert bases carried across 256 iterations. ~8192 items total.
// ---------------------------------------------------------------------------
__global__ __launch_bounds__(32)
void router_rank_scatter(const int* __restrict__ tk_idx, const float* __restrict__ tk_p,
                         float* __restrict__ out_used, float* __restrict__ out_cb,
                         float* __restrict__ out_mask, int N, int C) {
  const int lane = threadIdx.x;
  const unsigned lt_mask = (1u << lane) - 1u;   // lanes below me
  int base[8];
#pragma unroll
  for (int e = 0; e < 8; ++e) base[e] = 0;

  const int total = 2 * N;                       // flattened (k, n), k-major
  for (int i0 = 0; i0 < total; i0 += 32) {
    const int idx = i0 + lane;
    const int k = idx / N;
    const int n = idx - k * N;
    const int e = tk_idx[2 * n + k];
    const float p = tk_p[2 * n + k];

    int my_rank = 0x7FFFFFFF;
#pragma unroll
    for (int ex = 0; ex < 8; ++ex) {
      const unsigned bal = __builtin_amdgcn_ballot_w32(e == ex);
      if (e == ex) my_rank = base[ex] + __builtin_popcount(bal & lt_mask);
      base[ex] += __builtin_popcount(bal);       // uniform across lanes
    }
    if (my_rank < C) {
      const size_t off = ((size_t)n * 8 + e) * (size_t)C + (size_t)my_rank;
      out_cb[off]   = p;
      out_mask[off] = (p != 0.0f) ? 1.0f : 0.0f;
    }
  }
  if (lane == 0) {
#pragma unroll
    for (int e = 0; e < 8; ++e)
      out_used[e] = (float)(base[e] < C ? base[e] : C);
  }
}

// ---------------------------------------------------------------------------
extern "C" void kernel_launch(void* const* d_in, const int* in_sizes, int n_in,
                              void* d_out, int out_size, void* d_ws, size_t ws_size,
                              hipStream_t stream) {
  (void)n_in; (void)ws_size;
  const float* x  = (const float*)d_in[0];   // [2,2048,1024] fp32
  const float* wg = (const float*)d_in[1];   // [8,1024] fp32

  const int E = 8, K = 2;
  const int H = in_sizes[1] / E;             // 1024
  const int N = in_sizes[0] / H;             // 4096 tokens
  int capacity = (int)((double)K * 2.0 * (double)N / (double)E);  // floor
  if (capacity < 4) capacity = 4;            // MIN_CAPACITY

  float* out      = (float*)d_out;
  float* out_used = out;                                  // [E]
  float* out_cb   = out + E;                              // [N,E,C]
  float* out_mask = out_cb + (size_t)N * E * capacity;    // [N,E,C]

  int*   tk_idx = (int*)d_ws;                             // [N,2]
  float* tk_p   = (float*)((char*)d_ws + (size_t)N * 2 * sizeof(int));  // [N,2]

  // 1) zero the entire output (used_capacity + cb_weight + sec_mask)
  router_zero_fill<<<8192, 256, 0, stream>>>(out, (size_t)out_size);
  // 2) fp32 WMMA gating GEMM + top-2 + 2-way softmax
  router_gate_top2<<<N / 16, 32, 0, stream>>>(x, wg, tk_idx, tk_p, H);
  // 3) sequential capacity ranking + sparse scatter + used_capacity
  router_rank_scatter<<<1, 32, 0, stream>>>(tk_idx, tk_p, out_used, out_cb,
                                            out_mask, N, capacity);
}